// GCN_3Layer_37211596652978
// MI455X (gfx1250) — compile-verified
//
#include <hip/hip_runtime.h>

typedef __attribute__((ext_vector_type(2))) float v2f;
typedef __attribute__((ext_vector_type(8))) float v8f;

#define HID 64

// ---------- degree / normalization ----------
__global__ void k_deg_init(float* deg, int n) {
    int i = blockIdx.x * blockDim.x + threadIdx.x;
    if (i < n) deg[i] = 1.0f;                       // self loop contributes 1
}

__global__ void k_deg_edges(const int* __restrict__ dst, float* deg, int e) {
    int i = blockIdx.x * blockDim.x + threadIdx.x;
    if (i < e) atomicAdd(&deg[dst[i]], 1.0f);
}

__global__ void k_dinv(float* deg, int n) {          // deg -> deg^{-1/2} in place
    int i = blockIdx.x * blockDim.x + threadIdx.x;
    if (i < n) { float d = deg[i]; deg[i] = (d > 0.f) ? rsqrtf(d) : 0.f; }
}

// ---------- GEMM: C[n x 64] = (DO_RELU ? relu(A) : A)[n x K] @ W[K x 64] ----------
// One wave computes one 16x16 tile via V_WMMA_F32_16X16X4_F32; 4 waves/block
// cover the full 64-wide output. W staged in LDS. K and relu are compile-time
// so the k-loop fully unrolls into a straight WMMA chain with no selects.
template <int K, bool DO_RELU>
__global__ void k_gemm_wmma(const float* __restrict__ A, const float* __restrict__ W,
                            float* __restrict__ C, int n) {
    __shared__ float sW[K * HID];
    int tid = threadIdx.x;
    for (int i = tid; i < K * HID; i += 128) sW[i] = W[i];
    __syncthreads();

    int wave = tid >> 5;
    int lane = tid & 31;
    int half = lane >> 4;                           // 0: K={0,1}, 1: K={2,3}
    int l15  = lane & 15;
    int m0 = blockIdx.x * 16;
    int n0 = wave * 16;

    int arow = m0 + l15;
    if (arow >= n) arow = n - 1;                    // clamp loads; stores guarded
    const float* ap = A + (size_t)arow * K + half * 2;
    const float* bp = sW + n0 + l15 + half * 2 * HID;

    v8f c = {};
#pragma unroll
    for (int k = 0; k < K; k += 4) {
        v2f a, b;
        float ax = ap[k];
        float ay = ap[k + 1];
        if (DO_RELU) { ax = fmaxf(ax, 0.f); ay = fmaxf(ay, 0.f); }
        a[0] = ax; a[1] = ay;
        b[0] = bp[k * HID];                         // B row k+2*half
        b[1] = bp[k * HID + HID];                   // B row k+2*half+1
        c = __builtin_amdgcn_wmma_f32_16x16x4_f32(
                /*neg_a=*/false, a, /*neg_b=*/false, b,
                /*c_mod=*/(short)0, c, /*reuse_a=*/false, /*reuse_b=*/false);
    }

    int rbase = m0 + half * 8;                      // C: lanes 16-31 hold M+8
#pragma unroll
    for (int j = 0; j < 8; ++j) {
        int r = rbase + j;
        if (r < n) C[(size_t)r * HID + n0 + l15] = c[j];
    }
}

// ---------- aggregation: agg = D^-1/2 (A+I) D^-1/2 hW ----------
__global__ void k_selfinit(const float* __restrict__ hW, const float* __restrict__ dinv,
                           float* __restrict__ agg, int n) {
    int i = blockIdx.x * blockDim.x + threadIdx.x;
    if (i < n * HID) {
        int v = i >> 6;
        float di = dinv[v];
        agg[i] = hW[i] * di * di;                   // self-loop term == init
    }
}

__global__ void k_edge_scatter(const int* __restrict__ src, const int* __restrict__ dst,
                               const float* __restrict__ dinv, const float* __restrict__ hW,
                               float* __restrict__ agg, int e) {
    int i = blockIdx.x * blockDim.x + threadIdx.x;
    int total = e * HID;                            // 80M < 2^31
    if (i < total) {
        int ed = i >> 6, ch = i & 63;
        int s = src[ed], d = dst[ed];
        float nm = dinv[s] * dinv[d];
        atomicAdd(&agg[(size_t)d * HID + ch], hW[(size_t)s * HID + ch] * nm);
    }
}

// ---------- pooling + final linear ----------
__global__ void k_pool_zero(float* sums, float* cnt, int g) {
    int i = blockIdx.x * blockDim.x + threadIdx.x;
    if (i < g * HID) sums[i] = 0.f;
    if (i < g)       cnt[i]  = 0.f;
}

__global__ void k_pool_scatter(const float* __restrict__ h, const int* __restrict__ batch,
                               float* sums, float* cnt, int n) {
    int i = blockIdx.x * blockDim.x + threadIdx.x;
    if (i < n * HID) {
        int v = i >> 6, ch = i & 63;
        int g = batch[v];
        atomicAdd(&sums[(size_t)g * HID + ch], h[i]);
        if (ch == 0) atomicAdd(&cnt[g], 1.0f);
    }
}

__global__ void k_final(const float* __restrict__ sums, const float* __restrict__ cnt,
                        const float* __restrict__ Wl, const float* __restrict__ bl,
                        float* __restrict__ out, int g, int oc) {
    int i = blockIdx.x * blockDim.x + threadIdx.x;
    if (i < g * oc) {
        int gi = i / oc, o = i % oc;
        float inv = 1.0f / fmaxf(cnt[gi], 1.0f);
        float acc = bl[o];
        for (int c = 0; c < HID; ++c)
            acc += sums[(size_t)gi * HID + c] * inv * Wl[c * oc + o];
        out[i] = acc;
    }
}

extern "C" void kernel_launch(void* const* d_in, const int* in_sizes, int n_in,
                              void* d_out, int out_size, void* d_ws, size_t ws_size,
                              hipStream_t stream) {
    const float* x    = (const float*)d_in[0];
    const int*   eidx = (const int*)  d_in[1];
    const int*   batch= (const int*)  d_in[2];
    const float* W1   = (const float*)d_in[3];
    const float* W2   = (const float*)d_in[4];
    const float* W3   = (const float*)d_in[5];
    const float* Wl   = (const float*)d_in[6];
    const float* bl   = (const float*)d_in[7];

    const int INC = in_sizes[3] / HID;              // 128
    const int N   = in_sizes[0] / INC;              // 50000
    const int E   = in_sizes[1] / 2;                // 1.25M
    const int OC  = in_sizes[7];                    // 10
    const int G   = out_size / OC;                  // 500

    const int* src = eidx;
    const int* dst = eidx + E;

    // workspace carve-up (256B aligned): dinv | hW | agg | sums | cnt
    char* ws = (char*)d_ws; size_t off = 0;
    auto carve = [&](size_t bytes) {
        void* p = ws + off; off = (off + bytes + 255) & ~(size_t)255; return p;
    };
    float* dinv = (float*)carve((size_t)N * 4);
    float* hW   = (float*)carve((size_t)N * HID * 4);
    float* agg  = (float*)carve((size_t)N * HID * 4);
    float* sums = (float*)carve((size_t)G * HID * 4);
    float* cnt  = (float*)carve((size_t)G * 4);
    (void)ws_size; (void)n_in;

    const int B = 256;
    const int nB  = (N + B - 1) / B;
    const int eB  = (E + B - 1) / B;
    const int nhB = (N * HID + B - 1) / B;
    const int ehB = (E * HID + B - 1) / B;
    const int mt  = (N + 15) / 16;

    // normalization (once, shared by all 3 layers)
    k_deg_init <<<nB, B, 0, stream>>>(dinv, N);
    k_deg_edges<<<eB, B, 0, stream>>>(dst, dinv, E);
    k_dinv     <<<nB, B, 0, stream>>>(dinv, N);

    // layer 1: hW = x @ W1 ; agg1 = norm-agg(hW)
    k_gemm_wmma<128, false><<<mt, 128, 0, stream>>>(x, W1, hW, N);
    k_selfinit    <<<nhB, B, 0, stream>>>(hW, dinv, agg, N);
    k_edge_scatter<<<ehB, B, 0, stream>>>(src, dst, dinv, hW, agg, E);

    // layer 2: hW = relu(agg1) @ W2 ; agg2 = norm-agg(hW)
    k_gemm_wmma<64, true><<<mt, 128, 0, stream>>>(agg, W2, hW, N);
    k_selfinit    <<<nhB, B, 0, stream>>>(hW, dinv, agg, N);
    k_edge_scatter<<<ehB, B, 0, stream>>>(src, dst, dinv, hW, agg, E);

    // layer 3: hW = relu(agg2) @ W3 ; agg3 = norm-agg(hW) (no relu after)
    k_gemm_wmma<64, true><<<mt, 128, 0, stream>>>(agg, W3, hW, N);
    k_selfinit    <<<nhB, B, 0, stream>>>(hW, dinv, agg, N);
    k_edge_scatter<<<ehB, B, 0, stream>>>(src, dst, dinv, hW, agg, E);

    // global mean pool + final linear
    k_pool_zero   <<<(G * HID + B - 1) / B, B, 0, stream>>>(sums, cnt, G);
    k_pool_scatter<<<nhB, B, 0, stream>>>(agg, batch, sums, cnt, N);
    k_final       <<<(G * OC + B - 1) / B, B, 0, stream>>>(sums, cnt, Wl, bl,
                                                           (float*)d_out, G, OC);
}